// Capsule_25237227831320
// MI455X (gfx1250) — compile-verified
//
#include <hip/hip_runtime.h>
#include <math.h>

// Problem constants (from reference setup_inputs)
#define B_TOT 128
#define IN_N  2048
#define IN_D  8
#define CAPS  32
#define DIM   16
#define IC    32              // i's per workgroup chunk
#define NCHUNK (IN_N / IC)    // 64
#define NBT   (B_TOT / 16)    // 8 batch tiles of 16 (WMMA N)
#define JSTRIDE (IN_N * DIM * IN_D)   // W stride between capsules j (floats)

typedef float v2f __attribute__((ext_vector_type(2)));
typedef float v8f __attribute__((ext_vector_type(8)));

// D(16x16,f32) = A(16x4,f32) * B(4x16,f32) + C   — CDNA5 V_WMMA_F32_16X16X4_F32
__device__ __forceinline__ v8f wmma_k4(v2f a, v2f b, v8f c) {
    return __builtin_amdgcn_wmma_f32_16x16x4_f32(
        /*neg_a=*/false, a, /*neg_b=*/false, b,
        /*c_mod=*/(short)0, c, /*reuse_a=*/false, /*reuse_b=*/false);
}

__global__ void zero_f32(float* __restrict__ p, int n) {
    int i = blockIdx.x * blockDim.x + threadIdx.x;
    if (i < n) p[i] = 0.0f;
}

// One routing round. hat^T tile per (j,i): C[M=d][N=b] = W[j,i,d,e] * x[b,i,e].
// Lane owns b = lane&15; C vgpr v owns d = v + 8*(lane>>4).
__global__ __launch_bounds__(256) void caps_round(
    const float* __restrict__ x,      // [B, IN_N, IN_D]
    const float* __restrict__ W,      // [CAPS, IN_N, DIM, IN_D]
    const float* __restrict__ acc,    // [B, CAPS, DIM] routing vector (out0 or out0+out1)
    float* __restrict__ s_glob,       // [B, CAPS, DIM], pre-zeroed
    int round)
{
    __shared__ float s_tile[CAPS * DIM * 16];    // 32 KB  [j][d][b]
    __shared__ float acc_tile[CAPS * DIM * 16];  // 32 KB  [j][d][b]

    const int tid  = threadIdx.x;
    const int wave = tid >> 5;
    const int lane = tid & 31;
    const int half = lane >> 4;     // K rows (k0,k0+1) and C rows d = v + 8*half
    const int l16  = lane & 15;     // batch within tile (N), also A row d for loads
    const int k0   = half ? 2 : 0;

    const int i0 = blockIdx.x * IC;
    const int b0 = blockIdx.y * 16;

    for (int idx = tid; idx < CAPS * DIM * 16; idx += 256) s_tile[idx] = 0.0f;
    if (round > 0) {
        // stage acc as [j][d][b]; read side coalesced over (j,d)
        for (int idx = tid; idx < CAPS * DIM * 16; idx += 256) {
            const int b = idx >> 9;           // idx = b*512 + j*16 + d
            const int j = (idx >> 4) & 31;
            const int d = idx & 15;
            acc_tile[(j * DIM + d) * 16 + b] =
                acc[(((size_t)(b0 + b)) * CAPS + j) * DIM + d];
        }
    }
    __syncthreads();

    // Single divergent base for all LDS tile accesses: element (d=8*half, b=l16)
    float* const st = s_tile + 128 * half + l16;
    const float* const at = acc_tile + 128 * half + l16;

#pragma unroll 1
    for (int t = 0; t < IC / 8; ++t) {
        const int i = i0 + wave + t * 8;

        // B fragment (x): lane holds col N = l16 (batch), K pairs (k0,k0+1)/(4+k0,5+k0)
        const float* xp = x + ((size_t)(b0 + l16) * IN_N + i) * IN_D;
        const v2f xb_lo = *(const v2f*)(xp + k0);
        const v2f xb_hi = *(const v2f*)(xp + 4 + k0);

        // Single divergent 32-bit offset for all W loads (uniform j base goes to SGPRs)
        const int woff = (i * DIM + l16) * IN_D + k0;

        float lg[CAPS];   // per-lane coefficients for b = l16 (registers)

        if (round > 0) {
            // ---- pass 1: logits[b, j] = sum_d hat[b,d] * acc[b,j,d] ----
#pragma unroll
            for (int j = 0; j < CAPS; ++j) {
                const float* wj = W + (size_t)j * JSTRIDE;   // uniform base (SGPR)
                const v2f wa_lo = *(const v2f*)(wj + woff);
                const v2f wa_hi = *(const v2f*)(wj + woff + 4);
                v8f c = {0.f,0.f,0.f,0.f,0.f,0.f,0.f,0.f};
                c = wmma_k4(wa_lo, xb_lo, c);
                c = wmma_k4(wa_hi, xb_hi, c);
                float p = 0.0f;
#pragma unroll
                for (int v = 0; v < 8; ++v) {
                    p += c[v] * at[j * 256 + v * 16];   // acc[b=l16, j, d=v+8*half]
                }
                p += __shfl_xor(p, 16);   // combine d-halves; replicated to all lanes
                lg[j] = p;
            }
            // ---- per-lane softmax over j (b = l16) ----
            float mx = lg[0];
#pragma unroll
            for (int j = 1; j < CAPS; ++j) mx = fmaxf(mx, lg[j]);
            float sum = 0.0f;
#pragma unroll
            for (int j = 0; j < CAPS; ++j) { lg[j] = __expf(lg[j] - mx); sum += lg[j]; }
            const float inv = 1.0f / sum;
#pragma unroll
            for (int j = 0; j < CAPS; ++j) lg[j] *= inv;
        } else {
#pragma unroll
            for (int j = 0; j < CAPS; ++j) lg[j] = 1.0f / (float)CAPS;
        }

        // ---- pass 2: s_tile[j,d,b] += c[b,j] * hat[b,d] (recompute hat^T) ----
#pragma unroll
        for (int j = 0; j < CAPS; ++j) {
            const float* wj = W + (size_t)j * JSTRIDE;   // uniform base (SGPR)
            const v2f wa_lo = *(const v2f*)(wj + woff);
            const v2f wa_hi = *(const v2f*)(wj + woff + 4);
            v8f c = {0.f,0.f,0.f,0.f,0.f,0.f,0.f,0.f};
            c = wmma_k4(wa_lo, xb_lo, c);
            c = wmma_k4(wa_hi, xb_hi, c);
            const float cw = lg[j];
#pragma unroll
            for (int v = 0; v < 8; ++v) {
                atomicAdd(st + j * 256 + v * 16, c[v] * cw);   // conflict-free ds_add_f32
            }
        }
    }

    __syncthreads();
    // flush partial s to global with f32 atomics: s_tile[j][d][b] -> s[b0+b, j, d]
    for (int idx = tid; idx < CAPS * DIM * 16; idx += 256) {
        const int j = idx >> 8;            // idx = j*256 + d*16 + b
        const int d = (idx >> 4) & 15;
        const int b = idx & 15;
        atomicAdd(&s_glob[(((size_t)(b0 + b)) * CAPS + j) * DIM + d], s_tile[idx]);
    }
}

// squash: v = ||s||^2/(1+||s||^2) * s/sqrt(||s||^2+EPS); accumulate flag adds into out
__global__ void squash_kernel(const float* __restrict__ s,
                              float* __restrict__ out, int accumulate) {
    const int idx = blockIdx.x * blockDim.x + threadIdx.x;   // (b*CAPS + j)
    if (idx >= B_TOT * CAPS) return;
    const float* sp = s + (size_t)idx * DIM;
    float v[DIM];
    float sq = 0.0f;
#pragma unroll
    for (int d = 0; d < DIM; ++d) { v[d] = sp[d]; sq += v[d] * v[d]; }
    const float scale = sq / (1.0f + sq) / sqrtf(sq + 1e-7f);
#pragma unroll
    for (int d = 0; d < DIM; ++d) {
        const float o = scale * v[d];
        if (accumulate) out[(size_t)idx * DIM + d] += o;
        else            out[(size_t)idx * DIM + d]  = o;
    }
}

extern "C" void kernel_launch(void* const* d_in, const int* in_sizes, int n_in,
                              void* d_out, int out_size, void* d_ws, size_t ws_size,
                              hipStream_t stream) {
    (void)in_sizes; (void)n_in; (void)out_size; (void)ws_size;
    const float* x = (const float*)d_in[0];   // [128, 2048, 8]
    const float* W = (const float*)d_in[1];   // [32, 2048, 16, 8]
    float* out = (float*)d_out;               // [128, 32, 16]; doubles as routing accumulator
    float* s   = (float*)d_ws;                // [128, 32, 16] scratch (256 KB)

    const int NS = B_TOT * CAPS * DIM;        // 65536
    for (int r = 0; r < 3; ++r) {
        zero_f32<<<(NS + 255) / 256, 256, 0, stream>>>(s, NS);
        caps_round<<<dim3(NCHUNK, NBT), 256, 0, stream>>>(x, W, out, s, r);
        // r0: out = squash(s0); r1: out += squash(s1) (=> out0+out1); r2: out = final
        squash_kernel<<<(B_TOT * CAPS + 255) / 256, 256, 0, stream>>>(s, out,
                                                                     (r == 1) ? 1 : 0);
    }
}